// CapsuleModel2_72499047956924
// MI455X (gfx1250) — compile-verified
//
#include <hip/hip_runtime.h>
#include <hip/hip_bf16.h>

// ---------------------------------------------------------------------------
// CapsuleModel2: gather -> sorted-segment mean -> [4096,272]x[272,19] GEMM
// -> bias -> sigmoid.  MI455X (gfx1250, wave32).
//
// Shapes (fixed by reference): B=8,H=64,W=128 -> 65536 grid cells, D=272,
// NC=19, P=524288 points, NI=4096 instances.
// ---------------------------------------------------------------------------

#define D_DIM 272
#define NC_DIM 19
#define NI_DIM 4096

typedef __attribute__((ext_vector_type(2))) float v2f;
typedef __attribute__((ext_vector_type(8))) float v8f;

// ---------------------------------------------------------------------------
// Kernel 1: per-segment mean pool.
// segment_ids is sorted, so each block binary-searches its [lo,hi) point
// range, then accumulates the 272-dim rows with fully coalesced loads
// (lane t reads float t of the row). Gather table (71 MB) lives in L2.
// ---------------------------------------------------------------------------
__global__ void __launch_bounds__(256)
seg_mean_kernel(const float* __restrict__ caps,       // [65536, 272]
                const int* __restrict__ point_idx,    // [P]
                const int* __restrict__ seg_ids,      // [P] sorted
                float* __restrict__ pooled,           // [NI, 272] (workspace)
                int P) {
  const int seg = blockIdx.x;
  const int tid = threadIdx.x;

  // lower_bound(seg)
  int lo = 0, hi = P;
  while (lo < hi) {
    int mid = (lo + hi) >> 1;
    if (seg_ids[mid] < seg) lo = mid + 1; else hi = mid;
  }
  const int start = lo;
  // lower_bound(seg+1)
  hi = P;
  while (lo < hi) {
    int mid = (lo + hi) >> 1;
    if (seg_ids[mid] < seg + 1) lo = mid + 1; else hi = mid;
  }
  const int end = lo;

  float acc0 = 0.0f;
  float acc1 = 0.0f;  // dims 256..271, handled by tid < 16

  for (int p = start; p < end; ++p) {
    const long base = (long)point_idx[p] * D_DIM;
    if (p + 1 < end) {
      // Hide L2 gather latency: global_prefetch_b8 of the next row.
      __builtin_prefetch(&caps[(long)point_idx[p + 1] * D_DIM], 0, 0);
    }
    acc0 += caps[base + tid];
    if (tid < (D_DIM - 256)) acc1 += caps[base + 256 + tid];
  }

  const float cnt = (float)(end - start);
  const float inv = 1.0f / fmaxf(cnt, 1.0f);
  pooled[(long)seg * D_DIM + tid] = acc0 * inv;
  if (tid < (D_DIM - 256)) pooled[(long)seg * D_DIM + 256 + tid] = acc1 * inv;
}

// ---------------------------------------------------------------------------
// Kernel 2: out = sigmoid(pooled @ W + b) via V_WMMA_F32_16X16X4_F32.
// One wave (32 lanes) computes a 16x16 output tile.
//   M = 4096 -> 256 tiles; N = 19 -> 2 tiles (cols 19..31 zero-padded);
//   K = 272  -> 68 steps of K=4.
//
// W is staged into LDS once per block, zero-padded to 32 columns and laid
// out as lds_B[K/2][col][K&1] so each lane's B fragment (K pair {ak,ak+1}
// for its column) is one aligned ds_load_b64 with NO predication in the
// K loop (EXEC stays all-1s around every WMMA).
//
// A 16x4 f32 fragment: lanes 0-15 carry K={0,1}, lanes 16-31 carry K={2,3}.
// B 4x16 f32 fragment: N striped over lanes 0-15 / 16-31 the same way.
// C/D: 8 VGPRs, lane half selects M row group per ISA layout.
// ---------------------------------------------------------------------------
__global__ void __launch_bounds__(256)
gemm_sigmoid_wmma(const float* __restrict__ pooled,  // [NI, 272]
                  const float* __restrict__ Wmat,    // [272, 19] row-major
                  const float* __restrict__ bias,    // [19]
                  float* __restrict__ out) {         // [NI, 19]
  __shared__ float lds_B[(D_DIM / 2) * 64];  // [136][32][2] = 34816 B
  __shared__ float lds_bias[32];

  // Cooperative, zero-padded stage of W: element i = (kk, col, parity).
  for (int i = threadIdx.x; i < (D_DIM / 2) * 64; i += 256) {
    const int kk  = i >> 6;        // K/2
    const int rem = i & 63;
    const int n   = rem >> 1;      // column 0..31
    const int par = rem & 1;       // K parity
    const int K   = kk * 2 + par;
    lds_B[i] = (n < NC_DIM) ? Wmat[K * NC_DIM + n] : 0.0f;
  }
  if (threadIdx.x < 32)
    lds_bias[threadIdx.x] = (threadIdx.x < NC_DIM) ? bias[threadIdx.x] : 0.0f;
  __syncthreads();

  const int lane = threadIdx.x & 31;
  const int wave = threadIdx.x >> 5;
  const int tile = blockIdx.x * (blockDim.x >> 5) + wave;  // 0..511

  const int mtile = tile >> 1;        // 0..255
  const int ntile = tile & 1;         // 0..1
  const int m_base = mtile * 16;
  const int n_base = ntile * 16;

  const int lhalf = lane >> 4;        // selects K pair {0,1} vs {2,3}
  const int lmod  = lane & 15;        // A row (M) or B/C column (N)

  const int bn = n_base + lmod;       // LDS column 0..31 (already padded)

  v8f c = {0.f, 0.f, 0.f, 0.f, 0.f, 0.f, 0.f, 0.f};

  const long arow_off = (long)(m_base + lmod) * D_DIM;

#pragma unroll 4
  for (int k = 0; k < D_DIM; k += 4) {
    const int ak = k + 2 * lhalf;     // even

    // A fragment: one global_load_b64 of this lane's K pair.
    const v2f a = *(const v2f*)&pooled[arow_off + ak];

    // B fragment: one ds_load_b64, unconditional (padding handled at stage).
    const v2f bf = *(const v2f*)&lds_B[(ak >> 1) * 64 + bn * 2];

    // emits: v_wmma_f32_16x16x4_f32 (uniform control flow, EXEC all 1s)
    c = __builtin_amdgcn_wmma_f32_16x16x4_f32(
        /*neg_a=*/false, a, /*neg_b=*/false, bf,
        /*c_mod=*/(short)0, c, /*reuse_a=*/false, /*reuse_b=*/false);
  }

  // Epilogue: bias + sigmoid, store the 19 valid columns.
  const float bv = lds_bias[bn];
  const bool ncol_valid = (bn < NC_DIM);
#pragma unroll
  for (int i = 0; i < 8; ++i) {
    const int m = m_base + i + 8 * lhalf;    // C/D VGPR i -> rows i / i+8
    if (ncol_valid) {
      const float x = c[i] + bv;
      out[(long)m * NC_DIM + bn] = 1.0f / (1.0f + __expf(-x));
    }
  }
}

// ---------------------------------------------------------------------------
extern "C" void kernel_launch(void* const* d_in, const int* in_sizes, int n_in,
                              void* d_out, int out_size, void* d_ws, size_t ws_size,
                              hipStream_t stream) {
  const float* caps      = (const float*)d_in[0];  // [8,64,128,272]
  const float* Wmat      = (const float*)d_in[1];  // [272,19]
  const float* bias      = (const float*)d_in[2];  // [19]
  const int*   point_idx = (const int*)d_in[3];    // [P]
  const int*   seg_ids   = (const int*)d_in[4];    // [P] sorted
  const int P = in_sizes[3];

  float* pooled = (float*)d_ws;                    // NI*272 floats = 4.46 MB

  // 4096 segments, one block each; 256 threads cover the 272 feature dims.
  seg_mean_kernel<<<NI_DIM, 256, 0, stream>>>(caps, point_idx, seg_ids, pooled, P);

  // 512 wave-tiles (256 M-tiles x 2 N-tiles), 8 waves per 256-thread block.
  gemm_sigmoid_wmma<<<64, 256, 0, stream>>>(pooled, Wmat, bias, (float*)d_out);
}